// MultiRigidSidechain_13675175870802
// MI455X (gfx1250) — compile-verified
//
#include <hip/hip_runtime.h>
#include <hip/hip_bf16.h>

typedef float v2f __attribute__((ext_vector_type(2)));
typedef float v8f __attribute__((ext_vector_type(8)));

#define C_DIM   128
#define ROWS    32           // rows per workgroup (two 16-row WMMA tiles per wave)
#define LDS_W   132          // 128 + 4 pad: conflict-free column reads
#define R_TOT   65536        // B*N
#define OFF_ANG 0
#define OFF_UNN 917504       // 65536*14
#define OFF_POS 1835008      // 2*65536*14
#define OFF_ATG 4587520      // OFF_POS + 65536*42

// Single-instruction ReLU: v_med3_f32(x, 0, +inf) clamps to [0, inf)
__device__ __forceinline__ float relu1(float x) {
  return __builtin_amdgcn_fmed3f(x, 0.0f, __builtin_inff());
}

// ---- gfx1250 async global->LDS copy (ASYNCcnt-tracked, ISA §10 opcode 98) ----
__device__ __forceinline__ void async_copy_b128(const float* __restrict__ gbase,
                                                unsigned goff_bytes,
                                                unsigned lds_off_bytes) {
  asm volatile("global_load_async_to_lds_b128 %0, %1, %2"
               :
               : "v"(lds_off_bytes), "v"(goff_bytes), "s"(gbase)
               : "memory");
}

__device__ __forceinline__ void async_wait_all() {
  asm volatile("s_wait_asynccnt 0" ::: "memory");
}

// Preload this wave's 128x16 weight slice (row-major W[k][n], ldw=128) into regs.
__device__ __forceinline__ void load_btile(const float* __restrict__ W, int col, int kk,
                                           v2f* __restrict__ bfrag) {
#pragma unroll
  for (int ks = 0; ks < 32; ++ks) {
    const int k = ks * 4 + kk;
    bfrag[ks].x = W[k * C_DIM + col];
    bfrag[ks].y = W[(k + 1) * C_DIM + col];
  }
}

// Two independent 16x16 row tiles (rows m and m+16) against one B slice.
// acc{0,1} += (relu?)A @ B, K=128 via 2x32 wmma_f32_16x16x4_f32.
template <bool RELU>
__device__ __forceinline__ void wmma_k128x2(const float (*__restrict__ A)[LDS_W], int m, int kk,
                                            const v2f* __restrict__ bfrag,
                                            v8f& acc0, v8f& acc1) {
#pragma unroll
  for (int ks = 0; ks < 32; ++ks) {
    const int k = ks * 4 + kk;
    v2f a0, a1;
    a0.x = A[m][k];
    a0.y = A[m][k + 1];
    a1.x = A[m + 16][k];
    a1.y = A[m + 16][k + 1];
    if (RELU) {
      a0.x = relu1(a0.x); a0.y = relu1(a0.y);
      a1.x = relu1(a1.x); a1.y = relu1(a1.y);
    }
    acc0 = __builtin_amdgcn_wmma_f32_16x16x4_f32(false, a0, false, bfrag[ks],
                                                 (short)0, acc0, false, false);
    acc1 = __builtin_amdgcn_wmma_f32_16x16x4_f32(false, a1, false, bfrag[ks],
                                                 (short)0, acc1, false, false);
  }
}

__global__ __launch_bounds__(256)
void mrs_mlp_kernel(const float* __restrict__ act, const float* __restrict__ init_act,
                    const float* __restrict__ paw, const float* __restrict__ pab,
                    const float* __restrict__ piw, const float* __restrict__ pib,
                    const float* __restrict__ bw1, const float* __restrict__ bb1,
                    const float* __restrict__ bw2, const float* __restrict__ bb2,
                    const float* __restrict__ outw, const float* __restrict__ outb,
                    float* __restrict__ out) {
  __shared__ float Asm[ROWS][LDS_W];   // raw act (relu applied in fragment build)
  __shared__ float Hsm[ROWS][LDS_W];   // raw init_act, then resnet h (pre-relu'd)
  __shared__ float Xsm[ROWS][LDS_W];   // running x
  __shared__ float Usm[ROWS][14];      // unnorm head

  const int tid  = threadIdx.x;
  const int row0 = blockIdx.x * ROWS;

  // --- async DMA of both activation stripes into LDS (16 KB each) ---
  {
    const float* actBase  = act      + (size_t)row0 * C_DIM;
    const float* initBase = init_act + (size_t)row0 * C_DIM;
    for (int t = tid; t < (ROWS * C_DIM) / 4; t += 256) {   // b128 chunks
      const int e = t * 4;
      const int r = e >> 7, c = e & 127;
      const unsigned goff = (unsigned)t * 16u;
      async_copy_b128(actBase,  goff, (unsigned)(size_t)&Asm[r][c]);
      async_copy_b128(initBase, goff, (unsigned)(size_t)&Hsm[r][c]);
    }
    async_wait_all();
  }
  __syncthreads();

  const int lane    = tid & 31;
  const int wave    = tid >> 5;
  const int col     = wave * 16 + (lane & 15);
  const int m       = lane & 15;
  const int kk      = (lane >> 4) * 2;
  const int rowHalf = (lane >> 4) * 8;

  v2f bfrag[32];
  v8f xacc0, xacc1;
  {
    const float b0 = pab[col] + pib[col];
#pragma unroll
    for (int e = 0; e < 8; ++e) { xacc0[e] = b0; xacc1[e] = b0; }
  }
  load_btile(paw, col, kk, bfrag);
  wmma_k128x2<true>(Asm, m, kk, bfrag, xacc0, xacc1);   // relu(act)
  load_btile(piw, col, kk, bfrag);
  wmma_k128x2<true>(Hsm, m, kk, bfrag, xacc0, xacc1);   // relu(init_act)

#pragma unroll
  for (int r = 0; r < 8; ++r) {
    Xsm[r + rowHalf][col]      = xacc0[r];
    Xsm[16 + r + rowHalf][col] = xacc1[r];
  }
  __syncthreads();

  for (int blk = 0; blk < 2; ++blk) {
    const float* w1 = bw1 + blk * C_DIM * C_DIM;
    const float* w2 = bw2 + blk * C_DIM * C_DIM;
    const float b1v = bb1[blk * C_DIM + col];
    const float b2v = bb2[blk * C_DIM + col];

    v8f h0, h1;
#pragma unroll
    for (int e = 0; e < 8; ++e) { h0[e] = b1v; h1[e] = b1v; }
    load_btile(w1, col, kk, bfrag);
    wmma_k128x2<true>(Xsm, m, kk, bfrag, h0, h1);       // relu(x) on the fly

#pragma unroll
    for (int r = 0; r < 8; ++r) {
      Hsm[r + rowHalf][col]      = relu1(h0[r]);
      Hsm[16 + r + rowHalf][col] = relu1(h1[r]);
    }
    __syncthreads();                                    // h ready; X reads done

    v8f a0, a1;
#pragma unroll
    for (int e = 0; e < 8; ++e) { a0[e] = xacc0[e] + b2v; a1[e] = xacc1[e] + b2v; }
    load_btile(w2, col, kk, bfrag);
    wmma_k128x2<false>(Hsm, m, kk, bfrag, a0, a1);
    xacc0 = a0; xacc1 = a1;
#pragma unroll
    for (int r = 0; r < 8; ++r) {
      Xsm[r + rowHalf][col]      = xacc0[r];
      Xsm[16 + r + rowHalf][col] = xacc1[r];
    }
    __syncthreads();                                    // new x ready
  }

  // unnorm = relu(x) @ out_w[128,14] + out_b  (scalar head: 448 dots per block)
  for (int i = tid; i < ROWS * 14; i += 256) {
    const int r = i / 14, c = i % 14;
    float s = outb[c];
#pragma unroll 8
    for (int k = 0; k < C_DIM; ++k) s += relu1(Xsm[r][k]) * outw[k * 14 + c];
    Usm[r][c] = s;
    out[OFF_UNN + (size_t)(row0 + r) * 14 + c] = s;
  }
  __syncthreads();
  for (int i = tid; i < ROWS * 7; i += 256) {
    const int r = i / 7, j = i % 7;
    const float u0 = Usm[r][2 * j], u1 = Usm[r][2 * j + 1];
    const float inv = 1.0f / sqrtf(u0 * u0 + u1 * u1 + 1e-12f);
    const size_t base = OFF_ANG + (size_t)(row0 + r) * 14 + 2 * j;
    out[base]     = u0 * inv;
    out[base + 1] = u1 * inv;
  }
}

__global__ __launch_bounds__(256)
void mrs_frames_kernel(const int* __restrict__ seq, const float* __restrict__ backb,
                       const float* __restrict__ dframes, const int* __restrict__ gidx,
                       const float* __restrict__ amask, const float* __restrict__ lit,
                       float* __restrict__ out) {
  const int n = blockIdx.x * blockDim.x + threadIdx.x;
  if (n >= R_TOT) return;

  // affine mm: rows 0..2 x 4 cols, implicit bottom row [0,0,0,1]
  auto mm = [](const float* A, const float* Bm, float* Cm) {
#pragma unroll
    for (int r = 0; r < 3; ++r) {
#pragma unroll
      for (int c = 0; c < 4; ++c) {
        float v = A[r * 4 + 0] * Bm[0 * 4 + c] + A[r * 4 + 1] * Bm[1 * 4 + c] +
                  A[r * 4 + 2] * Bm[2 * 4 + c];
        if (c == 3) v += A[r * 4 + 3];
        Cm[r * 4 + c] = v;
      }
    }
  };

  const float* ang = out + OFF_ANG + (size_t)n * 14;
  float s8[8], c8[8];
  s8[0] = 0.f; c8[0] = 1.f;
#pragma unroll
  for (int j = 0; j < 7; ++j) { s8[j + 1] = ang[2 * j]; c8[j + 1] = ang[2 * j + 1]; }

  const int aa = seq[n];
  const float* df = dframes + (size_t)aa * 8 * 16;

  // all_frames[g] = default_frame @ x_rot(angle): mixes only columns 1 and 2
  float af[8][12];
#pragma unroll
  for (int g = 0; g < 8; ++g) {
    const float s = s8[g], c = c8[g];
    const float* d = df + g * 16;
#pragma unroll
    for (int r = 0; r < 3; ++r) {
      const float d0 = d[r * 4 + 0], d1 = d[r * 4 + 1], d2 = d[r * 4 + 2], d3 = d[r * 4 + 3];
      af[g][r * 4 + 0] = d0;
      af[g][r * 4 + 1] = c * d1 + s * d2;
      af[g][r * 4 + 2] = -s * d1 + c * d2;
      af[g][r * 4 + 3] = d3;
    }
  }

  // chi2 = af4@af5 -> af5 ; chi3 -> af6 ; chi4 -> af7
  float tmp[12];
  mm(af[4], af[5], tmp);
#pragma unroll
  for (int k = 0; k < 12; ++k) af[5][k] = tmp[k];
  mm(af[5], af[6], tmp);
#pragma unroll
  for (int k = 0; k < 12; ++k) af[6][k] = tmp[k];
  mm(af[6], af[7], tmp);
#pragma unroll
  for (int k = 0; k < 12; ++k) af[7][k] = tmp[k];

  const float* bb = backb + (size_t)n * 16;   // rows 0..2 used; row3 = [0,0,0,1]
  float atg[8][12];
  const size_t abase = OFF_ATG + (size_t)n * 128;
#pragma unroll
  for (int g = 0; g < 8; ++g) {
    mm(bb, af[g], atg[g]);
#pragma unroll
    for (int k = 0; k < 12; ++k) out[abase + g * 16 + k] = atg[g][k];
    out[abase + g * 16 + 12] = 0.f;
    out[abase + g * 16 + 13] = 0.f;
    out[abase + g * 16 + 14] = 0.f;
    out[abase + g * 16 + 15] = 1.f;
  }

  const size_t pbase = OFF_POS + (size_t)n * 42;
#pragma unroll
  for (int a = 0; a < 14; ++a) {
    const int g = gidx[aa * 14 + a];
    const float* L = lit + (size_t)(aa * 14 + a) * 3;
    const float lx = L[0], ly = L[1], lz = L[2];
    const float msk = amask[aa * 14 + a];
    const float* M = atg[g];
#pragma unroll
    for (int i = 0; i < 3; ++i) {
      out[pbase + a * 3 + i] =
          (M[i * 4 + 0] * lx + M[i * 4 + 1] * ly + M[i * 4 + 2] * lz + M[i * 4 + 3]) * msk;
    }
  }
}

extern "C" void kernel_launch(void* const* d_in, const int* in_sizes, int n_in,
                              void* d_out, int out_size, void* d_ws, size_t ws_size,
                              hipStream_t stream) {
  const int*   seq   = (const int*)d_in[0];
  const float* backb = (const float*)d_in[1];
  const float* act   = (const float*)d_in[2];
  const float* initv = (const float*)d_in[3];
  const float* paw   = (const float*)d_in[4];
  const float* pab   = (const float*)d_in[5];
  const float* piw   = (const float*)d_in[6];
  const float* pib   = (const float*)d_in[7];
  const float* bw1   = (const float*)d_in[8];
  const float* bb1   = (const float*)d_in[9];
  const float* bw2   = (const float*)d_in[10];
  const float* bb2   = (const float*)d_in[11];
  const float* outw  = (const float*)d_in[12];
  const float* outb  = (const float*)d_in[13];
  const float* dfr   = (const float*)d_in[14];
  const int*   gidx  = (const int*)d_in[15];
  const float* amask = (const float*)d_in[16];
  const float* lit   = (const float*)d_in[17];
  float* out = (float*)d_out;

  mrs_mlp_kernel<<<R_TOT / ROWS, 256, 0, stream>>>(act, initv, paw, pab, piw, pib,
                                                   bw1, bb1, bw2, bb2, outw, outb, out);
  mrs_frames_kernel<<<R_TOT / 256, 256, 0, stream>>>(seq, backb, dfr, gidx, amask, lit, out);
}